// CCL_50740743635433
// MI455X (gfx1250) — compile-verified
//
#include <hip/hip_runtime.h>
#include <math.h>

typedef __attribute__((ext_vector_type(2))) float v2f;
typedef __attribute__((ext_vector_type(8))) float v8f;

#define DIM     128
#define NCLASS  64
#define ALPHA_C 0.2f

// ws layout (floats):
//   [0, 64*128)       centroids (row-major [class][dim])
//   [8192, 8192+64)   csq  = ||centroid_c||^2
//   [8256, 8256+64)   cnt  = class counts (as float)
#define CENT_OFF 0
#define CSQ_OFF  (NCLASS * DIM)
#define CNT_OFF  (CSQ_OFF + NCLASS)

__global__ void ccl_zero_out(float* out) {
    if (threadIdx.x == 0 && blockIdx.x == 0) out[0] = 0.0f;
}

// One block per class, 128 threads (one per feature dim).
__global__ __launch_bounds__(DIM) void ccl_centroids(
    const float* __restrict__ preds, const int* __restrict__ labels,
    float* __restrict__ ws, int n_samples) {
    const int c = blockIdx.x;
    const int t = threadIdx.x;

    float s = 0.0f;
    int cnt = 0;
    for (int j = 0; j < n_samples; ++j) {
        const int lab = labels[j];          // uniform across block -> scalar load
        if (lab == c) { s += preds[(size_t)j * DIM + t]; ++cnt; }
    }
    const float cent = (cnt > 0) ? (s / (float)cnt) : 0.0f;
    ws[CENT_OFF + c * DIM + t] = cent;

    __shared__ float red[DIM];
    red[t] = cent * cent;
    __syncthreads();
    for (int off = DIM / 2; off > 0; off >>= 1) {
        if (t < off) red[t] += red[t + off];
        __syncthreads();
    }
    if (t == 0) {
        ws[CSQ_OFF + c] = red[0];
        ws[CNT_OFF + c] = (float)cnt;
    }
}

// One wave of 32 lanes handles a 16-row slab against all 64 classes.
// G = preds(16xK) @ cent^T(Kx64) via V_WMMA_F32_16X16X4_F32, K-loop step 4.
__global__ __launch_bounds__(32) void ccl_dist_loss(
    const float* __restrict__ preds, const int* __restrict__ labels,
    const float* __restrict__ ws, float* __restrict__ out, int n_samples) {
    const int lane    = threadIdx.x;        // wave32
    const int half    = lane >> 4;          // 0 or 1
    const int l15     = lane & 15;
    const int rowbase = blockIdx.x * 16;

    const float* cent = ws + CENT_OFF;
    const float* arow = preds + (size_t)(rowbase + l15) * DIM;

    v8f acc0 = {}, acc1 = {}, acc2 = {}, acc3 = {};
    float psq_part = 0.0f;

    for (int kk = 0; kk < DIM; kk += 4) {
        const int k = kk + 2 * half;        // this lane's k pair within the 4-slice
        // A fragment: row (rowbase + l15), elements k, k+1
        const v2f a = *(const v2f*)(arow + k);
        psq_part += a.x * a.x + a.y * a.y;
        // B fragments: class (t*16 + l15), elements k, k+1 (B = cent^T, KxN)
        const v2f b0 = *(const v2f*)(cent + (size_t)(0 * 16 + l15) * DIM + k);
        const v2f b1 = *(const v2f*)(cent + (size_t)(1 * 16 + l15) * DIM + k);
        const v2f b2 = *(const v2f*)(cent + (size_t)(2 * 16 + l15) * DIM + k);
        const v2f b3 = *(const v2f*)(cent + (size_t)(3 * 16 + l15) * DIM + k);
        acc0 = __builtin_amdgcn_wmma_f32_16x16x4_f32(false, a, false, b0, (short)0, acc0, false, false);
        acc1 = __builtin_amdgcn_wmma_f32_16x16x4_f32(false, a, false, b1, (short)0, acc1, false, false);
        acc2 = __builtin_amdgcn_wmma_f32_16x16x4_f32(false, a, false, b2, (short)0, acc2, false, false);
        acc3 = __builtin_amdgcn_wmma_f32_16x16x4_f32(false, a, false, b3, (short)0, acc3, false, false);
    }

    // ||p_m||^2 for m = rowbase + l15: combine the two k-halves (lanes l, l+16)
    psq_part += __shfl_xor(psq_part, 16, 32);

    // Per-class constants for this lane's 4 column positions
    float csq_l[4], present[4];
    for (int t = 0; t < 4; ++t) {
        const int n = t * 16 + l15;
        csq_l[t]   = ws[CSQ_OFF + n];
        present[t] = ws[CNT_OFF + n];
    }

    float wave_sum = 0.0f;
    for (int v = 0; v < 8; ++v) {
        // D layout: element (m, n) at vgpr (m&7), lane (m>>3)*16 + n
        const int m   = rowbase + v + 8 * half;
        const int lab = labels[m];
        const float pm = __shfl(psq_part, v + 8 * half, 32);  // ||p_m||^2

        float pos = 0.0f, neg = 1e30f;
        const float g[4] = { acc0[v], acc1[v], acc2[v], acc3[v] };
        for (int t = 0; t < 4; ++t) {
            const int n = t * 16 + l15;
            const float d2 = pm + csq_l[t] - 2.0f * g[t];
            const float d  = sqrtf(fmaxf(d2, 0.0f));
            if (n == lab)               pos = d;
            else if (present[t] > 0.0f) neg = fminf(neg, d);
        }
        // butterfly within each 16-lane group (rows m and m+8 reduce independently)
        for (int mask = 1; mask <= 8; mask <<= 1) {
            pos = fmaxf(pos, __shfl_xor(pos, mask, 32));
            neg = fminf(neg, __shfl_xor(neg, mask, 32));
        }
        if (l15 == 0) {
            const float x  = pos - neg + ALPHA_C;
            const float sp = fmaxf(x, 0.0f) + log1pf(expf(-fabsf(x)));  // stable softplus
            wave_sum += sp;
        }
    }

    wave_sum += __shfl_xor(wave_sum, 16, 32);   // lanes 0 and 16 hold the two halves
    if (lane == 0) atomicAdd(out, wave_sum / (float)n_samples);
}

extern "C" void kernel_launch(void* const* d_in, const int* in_sizes, int n_in,
                              void* d_out, int out_size, void* d_ws, size_t ws_size,
                              hipStream_t stream) {
    const float* preds  = (const float*)d_in[0];
    const int*   labels = (const int*)d_in[1];
    float* ws  = (float*)d_ws;
    float* out = (float*)d_out;

    const int n_samples = in_sizes[1];      // 8192 (multiple of 16)

    ccl_zero_out<<<1, 32, 0, stream>>>(out);
    ccl_centroids<<<NCLASS, DIM, 0, stream>>>(preds, labels, ws, n_samples);
    ccl_dist_loss<<<n_samples / 16, 32, 0, stream>>>(preds, labels, ws, out, n_samples);
}